// MultiLoraColumnParallelLinear_34007551050178
// MI455X (gfx1250) — compile-verified
//
#include <hip/hip_runtime.h>
#include <stdint.h>

// ---------------------------------------------------------------------------
// Multi-LoRA batched GEMM:  out[b] = X[b] (2048x4096) @ W[ids[b]] (4096x64)
// f32 in / f32 out.  HBM-bound (~47us floor @ 23.3 TB/s).
// CDNA5 path: async global->LDS double buffering (ASYNCcnt) for the X stream
// (non-temporal, read-once), register-pipelined transposed W staging, and
// full-precision V_WMMA_F32_16X16X4_F32 with b64 LDS fragments.
// ---------------------------------------------------------------------------

typedef __attribute__((ext_vector_type(2))) float v2f;
typedef __attribute__((ext_vector_type(8))) float v8f;

#define D_IN   4096
#define D_OUT  64
#define B_SZ   32
#define S_SZ   2048

#define KC     64            // K chunk staged in LDS per pipeline stage
#define NCHUNK (D_IN / KC)   // 64
#define MT     128           // rows of X per workgroup (8 waves x 16 rows)
#define XPAD   68            // X tile row stride (floats): 272B, 16B aligned;
                             // conflict-free b64 A-frag reads (bank pair 4L+c)
#define WPAD   68            // W^T tile row stride (floats): conflict-free b64
                             // B-frag reads (upper half-wave offset by +2 banks)

// Async 16B global->LDS copy, tracked by ASYNCcnt. Non-temporal: the X stream
// is read exactly once, keep L2 for the reused weight table.
__device__ __forceinline__ void async_ld_b128_nt(void* lds, const void* gptr) {
    uint32_t l = (uint32_t)(uintptr_t)lds;   // low 32 bits = LDS byte address
    asm volatile("global_load_async_to_lds_b128 %0, %1, off th:TH_LOAD_NT"
                 :: "v"(l), "v"(gptr) : "memory");
}

__global__ __launch_bounds__(256, 2)
void multilora_wmma_kernel(const float* __restrict__ x,
                           const int*   __restrict__ adapter_ids,
                           const float* __restrict__ w,
                           float*       __restrict__ out)
{
    __shared__ __align__(16) float Xl[2][MT * XPAD];      // 2 x 34816 B
    __shared__ __align__(16) float Wt[2][D_OUT * WPAD];   // 2 x 17408 B ([n][k])

    const int tid  = threadIdx.x;
    const int wave = tid >> 5;          // 0..7 (wave32)
    const int lane = tid & 31;

    const int mblocks = S_SZ / MT;                  // 16
    const int b  = blockIdx.x / mblocks;
    const int mb = blockIdx.x % mblocks;

    const int aid = adapter_ids[b];
    const float* xb = x   + (size_t)b   * S_SZ * D_IN + (size_t)mb * MT * D_IN;
    const float* wb = w   + (size_t)aid * D_IN * D_OUT;
    float*       ob = out + (size_t)b   * S_SZ * D_OUT + (size_t)mb * MT * D_OUT;

    // --- WMMA lane decomposition (16x16x4 f32) ---
    // A (16x4): lanes 0-15 hold M=lane, K={k0,k0+1}; lanes 16-31 M=lane-16, K={k0+2,k0+3}
    // B (4x16): lanes 0-15 hold N=lane, K={k0,k0+1}; lanes 16-31 N=lane-16, K={k0+2,k0+3}
    // C/D     : lanes 0-15 N=lane, M=vgpr;           lanes 16-31 N=lane-16, M=vgpr+8
    const int mrow  = wave * 16;
    const int laneM = lane & 15;
    const int klane = (lane >> 4) << 1;     // 0 for low half-wave, 2 for high

    v8f acc0 = {}, acc1 = {}, acc2 = {}, acc3 = {};

    // staging index precompute (256 threads)
    const int xv = tid & 15;            // float4 slot within a 64-float X row
    const int xr = tid >> 4;            // 0..15 row sub-index
    const int wn = tid & 63;            // W output-feature (column) index
    const int wk = tid >> 6;            // 0..3

    // ---- X staging: 8 async b128 copies per wave per chunk ----
    auto stage_x = [&](int buf, int kc) {
        float* Xd = Xl[buf];
        #pragma unroll
        for (int p = 0; p < 8; ++p) {
            const int r = p * 16 + xr;
            async_ld_b128_nt(&Xd[r * XPAD + xv * 4],
                             xb + (size_t)r * D_IN + kc + xv * 4);
        }
    };

    // ---- W staging: 16 coalesced b32 loads into registers (L2-resident),
    //      later scattered transposed into LDS ----
    float wreg[16];
    auto load_w = [&](int kc) {
        #pragma unroll
        for (int p = 0; p < 16; ++p) {
            const int kk = p * 4 + wk;                 // 0..63
            wreg[p] = wb[(size_t)(kc + kk) * D_OUT + wn];
        }
    };
    auto store_w = [&](int buf) {
        #pragma unroll
        for (int p = 0; p < 16; ++p) {
            const int kk = p * 4 + wk;
            Wt[buf][wn * WPAD + kk] = wreg[p];
        }
    };

    // ---- prologue: fill stage 0 ----
    stage_x(0, 0);
    load_w(0);
    store_w(0);

    for (int c = 0; c < NCHUNK; ++c) {
        const int cur = c & 1;

        if (c + 1 < NCHUNK) {
            stage_x(cur ^ 1, (c + 1) * KC);          // async prefetch of next X
            load_w((c + 1) * KC);                    // next W chunk in flight
            // 8 newest async issues = prefetch; oldest 8 (current) must land.
            asm volatile("s_wait_asynccnt 8" ::: "memory");
        } else {
            asm volatile("s_wait_asynccnt 0" ::: "memory");
        }
        __syncthreads();                             // chunk c fully visible

        // ---- compute: 16 k-steps x 4 N-tiles of v_wmma_f32_16x16x4_f32 ----
        const float* arow = &Xl[cur][(mrow + laneM) * XPAD + klane];
        const float* b0r  = &Wt[cur][( 0 + laneM) * WPAD + klane];
        const float* b1r  = &Wt[cur][(16 + laneM) * WPAD + klane];
        const float* b2r  = &Wt[cur][(32 + laneM) * WPAD + klane];
        const float* b3r  = &Wt[cur][(48 + laneM) * WPAD + klane];
        #pragma unroll
        for (int k0 = 0; k0 < KC; k0 += 4) {
            v2f a  = *(const v2f*)(arow + k0);
            v2f b0 = *(const v2f*)(b0r + k0);
            v2f b1 = *(const v2f*)(b1r + k0);
            v2f b2 = *(const v2f*)(b2r + k0);
            v2f b3 = *(const v2f*)(b3r + k0);
            acc0 = __builtin_amdgcn_wmma_f32_16x16x4_f32(false, a, false, b0, (short)0, acc0, false, false);
            acc1 = __builtin_amdgcn_wmma_f32_16x16x4_f32(false, a, false, b1, (short)0, acc1, false, false);
            acc2 = __builtin_amdgcn_wmma_f32_16x16x4_f32(false, a, false, b2, (short)0, acc2, false, false);
            acc3 = __builtin_amdgcn_wmma_f32_16x16x4_f32(false, a, false, b3, (short)0, acc3, false, false);
        }

        if (c + 1 < NCHUNK) {
            // Scatter next W chunk (transposed) into the other buffer; that
            // buffer is not read by any wave until after the next barrier.
            store_w(cur ^ 1);
        }
        __syncthreads();   // safe to overwrite buffer `cur` next iteration
    }

    // ---- write back C/D fragments ----
    // vgpr i, lane l: M = mrow + i + (l>=16 ? 8 : 0), N = (l&15) + 16*ntile
    const int mout = mrow + (lane >> 4) * 8;
    #pragma unroll
    for (int i = 0; i < 8; ++i) {
        float* orow = ob + (size_t)(mout + i) * D_OUT + laneM;
        orow[0]  = acc0[i];
        orow[16] = acc1[i];
        orow[32] = acc2[i];
        orow[48] = acc3[i];
    }
}

extern "C" void kernel_launch(void* const* d_in, const int* in_sizes, int n_in,
                              void* d_out, int out_size, void* d_ws, size_t ws_size,
                              hipStream_t stream) {
    const float* x   = (const float*)d_in[0];
    const int*   ids = (const int*)  d_in[1];
    const float* w   = (const float*)d_in[2];
    float*       o   = (float*)d_out;

    const dim3 grid(B_SZ * (S_SZ / MT));   // 32 * 16 = 512 workgroups
    const dim3 block(256);                 // 8 waves (wave32)
    multilora_wmma_kernel<<<grid, block, 0, stream>>>(x, ids, w, o);
}